// Dot_1743756722748
// MI455X (gfx1250) — compile-verified
//
#include <hip/hip_runtime.h>
#include <hip/hip_bf16.h>

typedef __attribute__((ext_vector_type(2))) float v2f;
typedef __attribute__((ext_vector_type(4))) float v4f;
typedef __attribute__((ext_vector_type(8))) float v8f;

#define DIM        128
#define ROWSTRIDE  132   // 128 + 4 pad floats -> bank stride 4, conflict-free b64 frag reads
#define TILE       16    // triplets per wave (one 16x16 WMMA C tile)
#define WPB        2     // waves per block
#define TPB        (WPB * TILE)

__global__ __launch_bounds__(WPB * 32) void dot_gather_wmma(
    const int* __restrict__ trip, const float* __restrict__ emb,
    float* __restrict__ out, int T)
{
  __shared__ float lds[WPB][2][TILE * ROWSTRIDE];

  const int lane = threadIdx.x & 31;
  const int wave = threadIdx.x >> 5;
  const int tileBase = (blockIdx.x * WPB + wave) * TILE;

  // lanes 0-15 fetch the LEFT index (col 0), lanes 16-31 the RIGHT index (col 2)
  // of triplet tileBase + (lane & 15). Clamp for the (non-existent) tail.
  int t = tileBase + (lane & 15);
  if (t >= T) t = T - 1;
  const int col = (lane < 16) ? 0 : 2;
  const int rowIdx = trip[t * 3 + col];

  float* ldsL = &lds[wave][0][0];
  float* ldsR = &lds[wave][1][0];

  // -------- Gather: 16 left + 16 right rows, one fully-coalesced 512B b128 load per row
  #pragma unroll
  for (int i = 0; i < TILE; ++i) {
    const int lrow = __shfl(rowIdx, i, 32);
    const int rrow = __shfl(rowIdx, i + 16, 32);
    const v4f lv = *(const v4f*)&emb[(size_t)lrow * DIM + lane * 4];
    const v4f rv = *(const v4f*)&emb[(size_t)rrow * DIM + lane * 4];
    *(v4f*)&ldsL[i * ROWSTRIDE + lane * 4] = lv;
    *(v4f*)&ldsR[i * ROWSTRIDE + lane * 4] = rv;
  }
  // Same-wave DS ops are in-order (DScnt); no barrier needed — waves are independent.

  const int m    = lane & 15;   // matrix row (A) / column (B) = triplet within tile
  const int half = lane >> 4;   // K sub-pair selector per the 16x16x4 f32 layout
  const int fragBase = m * ROWSTRIDE + half * 2;

#if __has_builtin(__builtin_amdgcn_wmma_f32_16x16x4_f32)
  // -------- C(16x16) = A(16x128) x B(128x16), 32 chained K=4 f32 WMMA steps
  v8f acc = {};
  #pragma unroll
  for (int j = 0; j < DIM / 4; ++j) {
    const v2f a = *(const v2f*)&ldsL[fragBase + j * 4];  // A[m][4j + 2*half .. +1]
    const v2f b = *(const v2f*)&ldsR[fragBase + j * 4];  // B[4j + 2*half][m] (= right_m[k])
    acc = __builtin_amdgcn_wmma_f32_16x16x4_f32(
        false, a, false, b, (short)0, acc, false, false);
  }

  // -------- Extract diagonal C[t][t] = dot(left_t, right_t)
  // C layout: lane n<16, VGPR r -> C[r][n]; lane n+16, VGPR r -> C[r+8][n]
  float* ex = ldsL;  // reuse staging LDS
  #pragma unroll
  for (int r = 0; r < 8; ++r) ex[lane * 8 + r] = acc[r];

  if (lane < 16) {
    const int srcLane = (lane < 8) ? lane : (lane + 16);
    const int srcV    = (lane < 8) ? lane : (lane - 8);
    const float val = ex[srcLane * 8 + srcV];
    const int tt = tileBase + lane;
    if (tt < T) out[tt] = val;
  }
#else
  // Fallback: plain f32 dot per triplet from the staged LDS rows
  if (lane < 16) {
    float s = 0.f;
    #pragma unroll 16
    for (int k = 0; k < DIM; ++k)
      s += ldsL[m * ROWSTRIDE + k] * ldsR[m * ROWSTRIDE + k];
    const int tt = tileBase + lane;
    if (tt < T) out[tt] = s;
  }
#endif
}

extern "C" void kernel_launch(void* const* d_in, const int* in_sizes, int n_in,
                              void* d_out, int out_size, void* d_ws, size_t ws_size,
                              hipStream_t stream) {
  const int*   trip = (const int*)d_in[0];    // triplets (T,3) int
  const float* emb  = (const float*)d_in[1];  // node_emb (N,128) f32
  float*       out  = (float*)d_out;          // scores (T,) f32
  const int T = in_sizes[0] / 3;
  const int blocks = (T + TPB - 1) / TPB;
  hipLaunchKernelGGL(dot_gather_wmma, dim3(blocks), dim3(WPB * 32), 0, stream,
                     trip, emb, out, T);
}